// Embedding_39573828665740
// MI455X (gfx1250) — compile-verified
//
#include <hip/hip_runtime.h>

// Embedding row-gather: out[r,:] = weight[token_ids[r],:]
// Pure data movement (~67 MB -> ~2.9 us at 23.3 TB/s HBM). CDNA5 async
// global<->LDS DMA path (ASYNCcnt) with GVS addressing: 64-bit base in
// SGPRs (kernel args), 32-bit per-lane offsets in VGPRs. Branch-free hot
// kernel over an exact grid; scalar tail kernel covers remainders (never
// launched for the harness geometry).

#define THREADS        256   // 8 waves (wave32)
#define ROWS_PER_WAVE  4
#define ROWS_PER_BLOCK 32    // 8 waves * 4 rows
#define DIM            512   // floats per row
#define DIM_BYTES      2048  // 512 f32
#define LANE_BYTES     16    // B128 per lane

__global__ __launch_bounds__(THREADS)
void embed_gather_async(const int* __restrict__ token_ids,
                        const float* __restrict__ weight,
                        float* __restrict__ out)
{
    extern __shared__ char smem[];
    const int lane    = threadIdx.x & 31;
    const int wave    = threadIdx.x >> 5;
    const int rowBase = blockIdx.x * ROWS_PER_BLOCK + wave * ROWS_PER_WAVE;

    // One broadcast 16B load fetches all 4 of this wave's tokens.
    const int4 t = *(const int4*)(token_ids + rowBase);
    int tok[ROWS_PER_WAVE] = {t.x, t.y, t.z, t.w};

    // Wave-private LDS region; lane's 16B slot within each 512B chunk.
    const unsigned ldsBase =
        (unsigned)(size_t)&smem[wave * ROWS_PER_WAVE * DIM_BYTES]
        + (unsigned)(lane * LANE_BYTES);
    const unsigned laneOff = (unsigned)(lane * LANE_BYTES);

    // ---- Phase 1: async gather weight rows -> LDS (GVS: saddr + u32 off) ----
    // weight table is 50257*2048 B ~= 103 MB, so tok*2048 fits in 32 bits.
    #pragma unroll
    for (int i = 0; i < ROWS_PER_WAVE; ++i) {
        const unsigned srcOff = (unsigned)tok[i] * DIM_BYTES + laneOff;
        const unsigned lds    = ldsBase + (unsigned)(i * DIM_BYTES);
        // IOFFSET is added to BOTH the LDS and the global address.
        asm volatile(
            "global_load_async_to_lds_b128 %0, %1, %2\n\t"
            "global_load_async_to_lds_b128 %0, %1, %2 offset:512\n\t"
            "global_load_async_to_lds_b128 %0, %1, %2 offset:1024\n\t"
            "global_load_async_to_lds_b128 %0, %1, %2 offset:1536"
            :: "v"(lds), "v"(srcOff), "s"(weight) : "memory");
    }

    // LDS regions are wave-private: wait only on our own async loads.
    asm volatile("s_wait_asynccnt 0" ::: "memory");

    // ---- Phase 2: async scatter LDS -> contiguous output rows ----
    // Output is 33.5 MB, so row*2048 fits in 32 bits. NT hint: write-once
    // stream; don't evict the weight table resident in the 192 MB L2.
    const unsigned dstBase = (unsigned)rowBase * DIM_BYTES + laneOff;
    #pragma unroll
    for (int i = 0; i < ROWS_PER_WAVE; ++i) {
        const unsigned dstOff = dstBase + (unsigned)(i * DIM_BYTES);
        const unsigned lds    = ldsBase + (unsigned)(i * DIM_BYTES);
        asm volatile(
            "global_store_async_from_lds_b128 %0, %1, %2 th:TH_STORE_NT\n\t"
            "global_store_async_from_lds_b128 %0, %1, %2 offset:512 th:TH_STORE_NT\n\t"
            "global_store_async_from_lds_b128 %0, %1, %2 offset:1024 th:TH_STORE_NT\n\t"
            "global_store_async_from_lds_b128 %0, %1, %2 offset:1536 th:TH_STORE_NT"
            :: "v"(dstOff), "v"(lds), "s"(out) : "memory");
    }
    // S_ENDPGM performs an implicit wait-idle, draining outstanding async stores.
}

// Generic remainder path (rows not covered by the exact async grid).
// For the harness geometry (16384 rows) this is never launched.
__global__ void embed_gather_tail(const int* __restrict__ token_ids,
                                  const float* __restrict__ weight,
                                  float* __restrict__ out,
                                  int startRow, int n_rows)
{
    const int row = startRow + blockIdx.x;
    if (row >= n_rows) return;
    const int tok = token_ids[row];
    const float4* __restrict__ src = (const float4*)(weight + (size_t)tok * DIM);
    float4* __restrict__ dst       = (float4*)(out + (size_t)row * DIM);
    for (int i = threadIdx.x; i < DIM / 4; i += blockDim.x)
        dst[i] = src[i];
}

extern "C" void kernel_launch(void* const* d_in, const int* in_sizes, int n_in,
                              void* d_out, int out_size, void* d_ws, size_t ws_size,
                              hipStream_t stream)
{
    const int*   token_ids = (const int*)d_in[0];      // [4*4096] int32
    const float* weight    = (const float*)d_in[1];    // [50257*512] f32
    float*       out       = (float*)d_out;            // [4*4096*512] f32

    const int n_rows      = in_sizes[0];               // 16384
    const int full_blocks = n_rows / ROWS_PER_BLOCK;   // 512 (exact grid)
    const int covered     = full_blocks * ROWS_PER_BLOCK;
    const int rem         = n_rows - covered;          // 0 for the harness

    if (full_blocks > 0) {
        const size_t lds = (size_t)ROWS_PER_BLOCK * (size_t)DIM_BYTES; // 64 KB
        embed_gather_async<<<full_blocks, THREADS, lds, stream>>>(token_ids, weight, out);
    }
    if (rem > 0) {
        embed_gather_tail<<<rem, 128, 0, stream>>>(token_ids, weight, out, covered, n_rows);
    }
}